// GCNGraphClassifier_541165879296
// MI455X (gfx1250) — compile-verified
//
#include <hip/hip_runtime.h>

#define N_NODES 50000
#define N_EDGES 800000
#define IN_CH   128
#define HID     64
#define NGRAPH  64
#define NCLASS  2

typedef __attribute__((ext_vector_type(2))) float v2f;
typedef __attribute__((ext_vector_type(8))) float v8f;

// ---------------- utility kernels ----------------

__global__ void zero_kernel(float* __restrict__ p, int n) {
  int i = blockIdx.x * blockDim.x + threadIdx.x;
  if (i < n) p[i] = 0.0f;
}

// deg[d] += 1 for each edge (self-loop +1 folded into dinv kernel)
__global__ void deg_kernel(const long long* __restrict__ dst, float* __restrict__ deg) {
  int e = blockIdx.x * blockDim.x + threadIdx.x;
  if (e < N_EDGES) atomicAdd(&deg[(int)dst[e]], 1.0f);
}

__global__ void dinv_kernel(const float* __restrict__ deg, float* __restrict__ dinv) {
  int n = blockIdx.x * blockDim.x + threadIdx.x;
  if (n < N_NODES) dinv[n] = rsqrtf(deg[n] + 1.0f);   // +1 self-loop
}

// ---------------- WMMA GEMM with fused self-loop epilogue ----------------
// T[N x 64] = A[N x K] * B[K x 64]   and   H[n][j] = T[n][j] * dinv[n]^2
// One 16x16 output tile per wave, 8 waves per 256-thread block -> 32 rows x 64 cols / block.
// f32 16x16x4 WMMA per CDNA5 ISA layouts:
//   A 16x4 : lane l -> m = l&15, khalf = (l>>4)*2 ; VGPR0 = A[m][k+kh], VGPR1 = A[m][k+kh+1]
//   B 4x16 : lane l -> n = l&15, khalf = (l>>4)*2 ; VGPR0 = B[k+kh][n], VGPR1 = B[k+kh+1][n]
//   C 16x16: VGPR v, lane l -> row = v + (l>>4)*8, col = l&15
template <int K>
__global__ __launch_bounds__(256) void gemm_wmma_kernel(const float* __restrict__ A,
                                                        const float* __restrict__ B,
                                                        const float* __restrict__ dinv,
                                                        float* __restrict__ T,
                                                        float* __restrict__ H) {
  const int wave = threadIdx.x >> 5;
  const int lane = threadIdx.x & 31;
  const int col0 = (wave & 3) * 16;
  const int row0 = (blockIdx.x * 2 + (wave >> 2)) * 16;
  if (row0 >= N_NODES) return;              // wave-uniform: EXEC stays all-ones for WMMA

  const int m  = lane & 15;                 // row within A tile / col within B tile
  const int kh = (lane >> 4) * 2;           // K half select
  const float* __restrict__ arow = A + (size_t)(row0 + m) * K;

  v8f c = {};
  for (int k0 = 0; k0 < K; k0 += 4) {
    v2f a, b;
    a.x = arow[k0 + kh];
    a.y = arow[k0 + kh + 1];
    b.x = B[(size_t)(k0 + kh) * HID + col0 + m];
    b.y = B[(size_t)(k0 + kh + 1) * HID + col0 + m];
    c = __builtin_amdgcn_wmma_f32_16x16x4_f32(false, a, false, b, (short)0, c, false, false);
  }

  const int mbase = row0 + (lane >> 4) * 8;
#pragma unroll
  for (int v = 0; v < 8; ++v) {
    const int r = mbase + v;
    const float di = dinv[r];
    const float val = c[v];
    T[(size_t)r * HID + col0 + m] = val;
    H[(size_t)r * HID + col0 + m] = val * di * di;   // self-loop contribution
  }
}

// ---------------- GCN edge aggregation (vectorized) ----------------
// h[dst][j..j+3] += t[src][j..j+3] * dinv[src]*dinv[dst] -- one thread per (edge, 4 channels)
__global__ __launch_bounds__(256) void scatter4_kernel(const long long* __restrict__ src,
                                                       const long long* __restrict__ dst,
                                                       const float* __restrict__ t,
                                                       const float* __restrict__ dinv,
                                                       float* __restrict__ h) {
  int idx = blockIdx.x * blockDim.x + threadIdx.x;   // E*16 = 12.8M < 2^31
  if (idx >= N_EDGES * 16) return;
  const int e  = idx >> 4;
  const int j4 = (idx & 15) * 4;
  const int s = (int)src[e];
  const int d = (int)dst[e];
  const float w = dinv[s] * dinv[d];
  const float4 tv = *(const float4*)(t + (size_t)s * HID + j4);
  float* hp = h + (size_t)d * HID + j4;
  atomicAdd(hp + 0, tv.x * w);
  atomicAdd(hp + 1, tv.y * w);
  atomicAdd(hp + 2, tv.z * w);
  atomicAdd(hp + 3, tv.w * w);
}

// h = relu(h + b), float4 vectorized (layer 1: result feeds next GEMM)
__global__ void bias_relu4_kernel(float* __restrict__ h, const float* __restrict__ b) {
  int idx = blockIdx.x * blockDim.x + threadIdx.x;   // N*16
  if (idx >= N_NODES * 16) return;
  const float4 bb = ((const float4*)b)[idx & 15];
  float4 v = ((const float4*)h)[idx];
  v.x = fmaxf(v.x + bb.x, 0.0f);
  v.y = fmaxf(v.y + bb.y, 0.0f);
  v.z = fmaxf(v.z + bb.z, 0.0f);
  v.w = fmaxf(v.w + bb.w, 0.0f);
  ((float4*)h)[idx] = v;
}

// Layer 2 epilogue fused with mean-pool accumulation: never materializes relu(h+b2).
// pacc[batch[n]][j..j+3] += relu(h[n][j..j+3] + b[j..j+3])
__global__ void bias_relu_pool4_kernel(const float* __restrict__ h, const float* __restrict__ b,
                                       const long long* __restrict__ batch,
                                       float* __restrict__ pacc) {
  int idx = blockIdx.x * blockDim.x + threadIdx.x;   // N*16
  if (idx >= N_NODES * 16) return;
  const int n   = idx >> 4;
  const int j4g = idx & 15;
  const float4 bb = ((const float4*)b)[j4g];
  const float4 v  = ((const float4*)h)[idx];
  const int g = (int)batch[n];
  float* pp = pacc + (size_t)g * HID + j4g * 4;
  atomicAdd(pp + 0, fmaxf(v.x + bb.x, 0.0f));
  atomicAdd(pp + 1, fmaxf(v.y + bb.y, 0.0f));
  atomicAdd(pp + 2, fmaxf(v.z + bb.z, 0.0f));
  atomicAdd(pp + 3, fmaxf(v.w + bb.w, 0.0f));
}

// ---------------- pooling + classifier ----------------

__global__ void count_kernel(const long long* __restrict__ batch, float* __restrict__ counts) {
  int n = blockIdx.x * blockDim.x + threadIdx.x;
  if (n < N_NODES) atomicAdd(&counts[(int)batch[n]], 1.0f);
}

__global__ void poolfin_kernel(const float* __restrict__ acc, const float* __restrict__ counts,
                               float* __restrict__ pooled_out) {
  int idx = blockIdx.x * blockDim.x + threadIdx.x;
  if (idx < NGRAPH * HID)
    pooled_out[idx] = acc[idx] / fmaxf(counts[idx >> 6], 1.0f);
}

// out[g][c] = bl[c] + sum_j pooled[g][j] * Wl[j][c]   (64x64 @ 64x2 -- tiny, scalar)
__global__ void classify_kernel(const float* __restrict__ pooled, const float* __restrict__ Wl,
                                const float* __restrict__ bl, float* __restrict__ out) {
  int t = blockIdx.x * blockDim.x + threadIdx.x;
  if (t >= NGRAPH * NCLASS) return;
  int g = t / NCLASS, c = t % NCLASS;
  float acc = bl[c];
#pragma unroll 8
  for (int j = 0; j < HID; ++j) acc += pooled[g * HID + j] * Wl[j * NCLASS + c];
  out[g * NCLASS + c] = acc;
}

// ---------------- launcher ----------------

extern "C" void kernel_launch(void* const* d_in, const int* in_sizes, int n_in,
                              void* d_out, int out_size, void* d_ws, size_t ws_size,
                              hipStream_t stream) {
  (void)in_sizes; (void)n_in; (void)out_size; (void)ws_size;

  const float*     x     = (const float*)d_in[0];          // [N, 128]
  const long long* ei    = (const long long*)d_in[1];      // [2, E] int64
  const long long* batch = (const long long*)d_in[2];      // [N] int64
  const float*     W1    = (const float*)d_in[3];          // [128, 64]
  const float*     b1    = (const float*)d_in[4];          // [64]
  const float*     W2    = (const float*)d_in[5];          // [64, 64]
  const float*     b2    = (const float*)d_in[6];          // [64]
  const float*     Wl    = (const float*)d_in[7];          // [64, 2]
  const float*     bl    = (const float*)d_in[8];          // [2]

  const long long* src = ei;
  const long long* dst = ei + N_EDGES;

  // workspace layout (floats)
  float* bufT   = (float*)d_ws;                   // [N*64] linear-transform output
  float* bufH   = bufT + (size_t)N_NODES * HID;   // [N*64] aggregated output
  float* deg    = bufH + (size_t)N_NODES * HID;   // [N]
  float* dinv   = deg + N_NODES;                  // [N]
  float* pacc   = dinv + N_NODES;                 // [G*64]
  float* counts = pacc + NGRAPH * HID;            // [G]

  float* out_logits = (float*)d_out;                       // [G*2]
  float* out_pooled = (float*)d_out + NGRAPH * NCLASS;     // [G*64]

  const int TPB = 256;
  const int zeroN = 2 * N_NODES + NGRAPH * HID + NGRAPH;   // deg,dinv,pacc,counts
  const int nv4 = N_NODES * 16;                            // float4 elems over [N,64]
  const int ev4 = N_EDGES * 16;                            // (edge, 4-chan) work items
  const int gemmBlocks = (N_NODES + 31) / 32;

  // degrees & normalization
  zero_kernel<<<(zeroN + TPB - 1) / TPB, TPB, 0, stream>>>(deg, zeroN);
  deg_kernel<<<(N_EDGES + TPB - 1) / TPB, TPB, 0, stream>>>(dst, deg);
  dinv_kernel<<<(N_NODES + TPB - 1) / TPB, TPB, 0, stream>>>(deg, dinv);

  // layer 1: T = x @ W1 (+fused H = T*dinv^2) ; H += edges ; H = relu(H + b1)
  gemm_wmma_kernel<IN_CH><<<gemmBlocks, TPB, 0, stream>>>(x, W1, dinv, bufT, bufH);
  scatter4_kernel<<<(ev4 + TPB - 1) / TPB, TPB, 0, stream>>>(src, dst, bufT, dinv, bufH);
  bias_relu4_kernel<<<(nv4 + TPB - 1) / TPB, TPB, 0, stream>>>(bufH, b1);

  // layer 2: T = H @ W2 (+fused H = T*dinv^2) ; H += edges ; pool(relu(H + b2))
  gemm_wmma_kernel<HID><<<gemmBlocks, TPB, 0, stream>>>(bufH, W2, dinv, bufT, bufH);
  scatter4_kernel<<<(ev4 + TPB - 1) / TPB, TPB, 0, stream>>>(src, dst, bufT, dinv, bufH);
  count_kernel<<<(N_NODES + TPB - 1) / TPB, TPB, 0, stream>>>(batch, counts);
  bias_relu_pool4_kernel<<<(nv4 + TPB - 1) / TPB, TPB, 0, stream>>>(bufH, b2, batch, pacc);

  // finalize pool + classifier head
  poolfin_kernel<<<(NGRAPH * HID + TPB - 1) / TPB, TPB, 0, stream>>>(pacc, counts, out_pooled);
  classify_kernel<<<1, 128, 0, stream>>>(out_pooled, Wl, bl, out_logits);
}